// MultiHeadAttentionLayerGrit_44057774522784
// MI455X (gfx1250) — compile-verified
//
#include <hip/hip_runtime.h>
#include <hip/hip_bf16.h>

#define N_NODES 40000
#define N_EDGES 640000
#define C_IN    64
#define NH      8
#define DH      16
#define HDIM    128   // NH * DH
#define N_TILES_NODE (N_NODES / 16)   // 2500

typedef __attribute__((ext_vector_type(2))) float v2f;
typedef __attribute__((ext_vector_type(8))) float v8f;
typedef unsigned int u32x4 __attribute__((ext_vector_type(4)));
typedef int          i32x8 __attribute__((ext_vector_type(8)));
typedef int          i32x4 __attribute__((ext_vector_type(4)));

// ---- order-preserving float <-> uint map for atomic segment-max ----
__device__ __forceinline__ unsigned f2ord(float f) {
    unsigned u = __float_as_uint(f);
    return (u & 0x80000000u) ? ~u : (u | 0x80000000u);
}
__device__ __forceinline__ float ord2f(unsigned u) {
    return (u & 0x80000000u) ? __uint_as_float(u & 0x7fffffffu)
                             : __uint_as_float(~u);
}

// ---------------------------------------------------------------------------
// TDM: DMA a contiguous run of n_dwords f32 from global into LDS.
// D# per CDNA5 ISA:
//  g0: [1:0]=count=1 | [63:32]=lds_addr | [120:64]=global_addr | [127:126]=type=2
//  g1: data_size=2(4B), tensor_dim0=tile_dim0=n_dwords, dims1=1, strides=n_dwords
//  g2/g3/g4: zero (tensor <= 2D)
// 6-arg builtin form (clang-23 / therock-10.0 toolchain).
// ---------------------------------------------------------------------------
__device__ __forceinline__ void tdm_load_1d(unsigned lds_off, const void* gptr,
                                            unsigned n_dwords)
{
    unsigned long long ga = (unsigned long long)(uintptr_t)gptr;
    u32x4 g0;
    g0[0] = 1u;                                                  // count=1
    g0[1] = lds_off;                                             // lds_addr (bytes)
    g0[2] = (unsigned)ga;                                        // global_addr[31:0]
    g0[3] = (unsigned)((ga >> 32) & 0x01FFFFFFu) | (2u << 30);   // addr[56:32] | type=2

    i32x8 g1;
    g1[0] = (int)(2u << 16);                                     // data_size = 4 bytes
    g1[1] = (int)((n_dwords & 0xFFFFu) << 16);                   // tensor_dim0[15:0]
    g1[2] = (int)((n_dwords >> 16) | (1u << 16));                // tensor_dim0[31:16] | tensor_dim1=1
    g1[3] = (int)((n_dwords & 0xFFFFu) << 16);                   // tile_dim0 (16-bit)
    g1[4] = 1;                                                   // tile_dim1=1, tile_dim2=0
    g1[5] = (int)n_dwords;                                       // tensor_dim0_stride[31:0]
    g1[6] = (int)((n_dwords & 0xFFFFu) << 16);                   // stride0[47:32]=0 | stride1[15:0]
    g1[7] = (int)(n_dwords >> 16);                               // stride1[47:16]
    i32x4 gz4 = {0, 0, 0, 0};
    i32x8 gz8 = {0, 0, 0, 0, 0, 0, 0, 0};
    __builtin_amdgcn_tensor_load_to_lds(g0, g1, gz4, gz4, gz8, 0);
}

// ---------------------------------------------------------------------------
// K0: init accumulators (must run every call; harness poisons ws once)
// ---------------------------------------------------------------------------
__global__ void grit_init(unsigned* __restrict__ m_ord, float* __restrict__ den,
                          float* __restrict__ node_acc, float* __restrict__ alpha)
{
    int i = blockIdx.x * blockDim.x + threadIdx.x;
    if (i < N_NODES * NH) {
        m_ord[i] = f2ord(-3.0e38f);
        den[i]   = 0.0f;
    }
    if (i < N_NODES * HDIM) {
        node_acc[i] = 0.0f;
        alpha[i]    = 0.0f;
    }
}

// ---------------------------------------------------------------------------
// K1: Qx/Kx/Vx = x @ W + b  via V_WMMA_F32_16X16X4_F32.
// 8 waves/block = 8 row-tiles; each of the 3 weight matrices is TDM-staged
// into one 32KB LDS buffer, amortizing weight reads 8x across waves.
// ---------------------------------------------------------------------------
__global__ __launch_bounds__(256)
void grit_node_gemm(const float* __restrict__ x,
                    const float* __restrict__ WQ, const float* __restrict__ bQ,
                    const float* __restrict__ WK, const float* __restrict__ bK,
                    const float* __restrict__ WV, const float* __restrict__ bV,
                    float* __restrict__ Qx, float* __restrict__ Kx, float* __restrict__ Vx)
{
    __shared__ float sW[C_IN * HDIM];            // 32 KB

    const int lane  = threadIdx.x & 31;
    const int wave  = threadIdx.x >> 5;
    const int half  = lane >> 4;
    const int n     = lane & 15;
    const int tile  = blockIdx.x * 8 + wave;
    const bool valid = (tile < N_TILES_NODE);
    const int row0  = tile * 16;

    v2f a[16];
    if (valid) {
        const float* xrow = x + (size_t)(row0 + n) * C_IN;
#pragma unroll
        for (int kk = 0; kk < 16; ++kk) {
            int k = kk * 4 + half * 2;
            a[kk].x = xrow[k];
            a[kk].y = xrow[k + 1];
        }
    }

    const float* Ws[3]  = {WQ, WK, WV};
    const float* bs[3]  = {bQ, bK, bV};
    float*       dst[3] = {Qx, Kx, Vx};
    const unsigned lds_off = (unsigned)(uintptr_t)(void*)&sW[0];

    for (int m = 0; m < 3; ++m) {
        if (wave == 0) {                          // one wave drives the TDM
            tdm_load_1d(lds_off, Ws[m], C_IN * HDIM);
            __builtin_amdgcn_s_wait_tensorcnt(0);
        }
        __syncthreads();
        if (valid) {
#pragma unroll
            for (int t = 0; t < 8; ++t) {
                const int col = t * 16 + n;
                v8f c = {};
#pragma unroll
                for (int kk = 0; kk < 16; ++kk) {
                    int k = kk * 4 + half * 2;
                    v2f b;
                    b.x = sW[(k    ) * HDIM + col];
                    b.y = sW[(k + 1) * HDIM + col];
                    c = __builtin_amdgcn_wmma_f32_16x16x4_f32(false, a[kk], false, b,
                                                              (short)0, c, false, false);
                }
                const float bias = bs[m][col];
#pragma unroll
                for (int p = 0; p < 8; ++p)
                    dst[m][(size_t)(row0 + half * 8 + p) * HDIM + col] = c[p] + bias;
            }
        }
        __syncthreads();                          // before next TDM overwrites sW
    }
}

// ---------------------------------------------------------------------------
// K2: edge GEMMs (Ew, Eb) + fused elementwise + edge_rep + per-head scores.
// 8 waves/block = 128 edges; both weight matrices TDM-staged to LDS (64 KB).
// Each 16-col tile is one head, so score s.A[:,h] reduces via shfl_xor
// inside each 16-lane half of the WMMA accumulator layout.
// ---------------------------------------------------------------------------
__global__ __launch_bounds__(256)
void grit_edge_gemm(const float* __restrict__ ea,
                    const float* __restrict__ WEw, const float* __restrict__ bEw,
                    const float* __restrict__ WEb, const float* __restrict__ bEb,
                    const int* __restrict__ srcI, const int* __restrict__ dstI,
                    const float* __restrict__ Qx, const float* __restrict__ Kx,
                    const float* __restrict__ Avec,      // [D, H]
                    float* __restrict__ edge_rep,        // [E, HD] (in d_out)
                    float* __restrict__ scal,            // [E, H]
                    unsigned* __restrict__ m_ord)        // [N, H]
{
    __shared__ float sWw[C_IN * HDIM];           // 32 KB
    __shared__ float sWb[C_IN * HDIM];           // 32 KB

    const int lane = threadIdx.x & 31;
    const int wave = threadIdx.x >> 5;
    const int half = lane >> 4;
    const int n    = lane & 15;
    const int row0 = (blockIdx.x * 8 + wave) * 16;

    if (wave == 0) {
        tdm_load_1d((unsigned)(uintptr_t)(void*)&sWw[0], WEw, C_IN * HDIM);
        tdm_load_1d((unsigned)(uintptr_t)(void*)&sWb[0], WEb, C_IN * HDIM);
        __builtin_amdgcn_s_wait_tensorcnt(0);
    }

    v2f a[16];
    {
        const float* arow = ea + (size_t)(row0 + n) * C_IN;
#pragma unroll
        for (int kk = 0; kk < 16; ++kk) {
            int k = kk * 4 + half * 2;
            a[kk].x = arow[k];
            a[kk].y = arow[k + 1];
        }
    }
    int srcs[8], dsts[8];
#pragma unroll
    for (int p = 0; p < 8; ++p) {
        srcs[p] = srcI[row0 + half * 8 + p];
        dsts[p] = dstI[row0 + half * 8 + p];
    }
    __syncthreads();                              // LDS weights visible

    for (int h = 0; h < NH; ++h) {                // col tile == head
        const int col = h * 16 + n;
        v8f cw = {}, cb = {};
#pragma unroll
        for (int kk = 0; kk < 16; ++kk) {
            int k = kk * 4 + half * 2;
            v2f bw, bb;
            bw.x = sWw[(k    ) * HDIM + col];
            bw.y = sWw[(k + 1) * HDIM + col];
            bb.x = sWb[(k    ) * HDIM + col];
            bb.y = sWb[(k + 1) * HDIM + col];
            cw = __builtin_amdgcn_wmma_f32_16x16x4_f32(false, a[kk], false, bw,
                                                       (short)0, cw, false, false);
            cb = __builtin_amdgcn_wmma_f32_16x16x4_f32(false, a[kk], false, bb,
                                                       (short)0, cb, false, false);
        }
        const float biasw = bEw[col];
        const float biasb = bEb[col];
        const float aw    = Avec[n * NH + h];     // A[d = n, h]

        float sc[8];
#pragma unroll
        for (int p = 0; p < 8; ++p) {
            float ew = cw[p] + biasw;
            float eb = cb[p] + biasb;
            float q  = Qx[(size_t)srcs[p] * HDIM + col];
            float kx = Kx[(size_t)dsts[p] * HDIM + col];
            float s  = (q + kx) * ew;
            s = copysignf(sqrtf(fabsf(s)), s);    // signed sqrt
            s = s + eb;
            s = fmaxf(s, 0.0f);                   // relu -> edge_rep
            edge_rep[(size_t)(row0 + half * 8 + p) * HDIM + col] = s;
            sc[p] = s * aw;
        }
#pragma unroll
        for (int off = 1; off < 16; off <<= 1) {  // reduce over d within half
#pragma unroll
            for (int p = 0; p < 8; ++p)
                sc[p] += __shfl_xor(sc[p], off, 32);
        }
        if (n == 0) {
#pragma unroll
            for (int p = 0; p < 8; ++p) {
                scal[(size_t)(row0 + half * 8 + p) * NH + h] = sc[p];
                atomicMax(&m_ord[dsts[p] * NH + h], f2ord(sc[p]));
            }
        }
    }
}

// ---------------------------------------------------------------------------
// K3: ex = exp(scal - m[dst]); den[dst] += ex    (scal overwritten in place)
// ---------------------------------------------------------------------------
__global__ void grit_softmax_num(const int* __restrict__ dstI,
                                 float* __restrict__ scal,
                                 const unsigned* __restrict__ m_ord,
                                 float* __restrict__ den)
{
    int i = blockIdx.x * blockDim.x + threadIdx.x;
    if (i >= N_EDGES * NH) return;
    int e = i / NH;
    int h = i - e * NH;
    int d = dstI[e];
    float mm = ord2f(m_ord[d * NH + h]);
    float ex = __expf(scal[i] - mm);
    scal[i] = ex;
    atomicAdd(&den[d * NH + h], ex);
}

// ---------------------------------------------------------------------------
// K4: scatter-add attn*Vx[src] -> node_acc ; attn*edge_rep -> alpha
// ---------------------------------------------------------------------------
__global__ void grit_scatter(const int* __restrict__ srcI, const int* __restrict__ dstI,
                             const float* __restrict__ ex, const float* __restrict__ den,
                             const float* __restrict__ Vx, const float* __restrict__ edge_rep,
                             float* __restrict__ node_acc, float* __restrict__ alpha)
{
    int i = blockIdx.x * blockDim.x + threadIdx.x;   // E*HD = 81.92M < 2^31
    if (i >= N_EDGES * HDIM) return;
    int e  = i >> 7;
    int hd = i & 127;
    int h  = hd >> 4;
    int d  = dstI[e];
    int s  = srcI[e];
    float attn = ex[e * NH + h] / (den[d * NH + h] + 1e-16f);
    atomicAdd(&node_acc[(size_t)d * HDIM + hd], attn * Vx[(size_t)s * HDIM + hd]);
    atomicAdd(&alpha[(size_t)d * HDIM + hd],    attn * edge_rep[(size_t)e * HDIM + hd]);
}

// ---------------------------------------------------------------------------
// K5: node = node_acc + einsum('nhd,dhc->nhc', alpha, E_v)
// ---------------------------------------------------------------------------
__global__ void grit_node_out(const float* __restrict__ node_acc,
                              const float* __restrict__ alpha,
                              const float* __restrict__ Ev,    // [D, H, D]
                              float* __restrict__ out_node)
{
    int i = blockIdx.x * blockDim.x + threadIdx.x;
    if (i >= N_NODES * HDIM) return;
    int nd = i >> 7;
    int hd = i & 127;
    int h  = hd >> 4;
    int c  = hd & 15;
    float acc = node_acc[i];
    const float* al = alpha + (size_t)nd * HDIM + h * 16;
#pragma unroll
    for (int d = 0; d < DH; ++d)
        acc += al[d] * Ev[(d * NH + h) * DH + c];
    out_node[i] = acc;
}

// ---------------------------------------------------------------------------
extern "C" void kernel_launch(void* const* d_in, const int* in_sizes, int n_in,
                              void* d_out, int out_size, void* d_ws, size_t ws_size,
                              hipStream_t stream)
{
    const float* x      = (const float*)d_in[0];
    const float* eattr  = (const float*)d_in[1];
    const int*   eidx   = (const int*)  d_in[2];   // [2, E]
    const float* WQ  = (const float*)d_in[3];  const float* bQ  = (const float*)d_in[4];
    const float* WK  = (const float*)d_in[5];  const float* bK  = (const float*)d_in[6];
    const float* WEw = (const float*)d_in[7];  const float* bEw = (const float*)d_in[8];
    const float* WEb = (const float*)d_in[9];  const float* bEb = (const float*)d_in[10];
    const float* WV  = (const float*)d_in[11]; const float* bV  = (const float*)d_in[12];
    const float* Ev  = (const float*)d_in[13];
    const float* A   = (const float*)d_in[14];

    const int* srcI = eidx;
    const int* dstI = eidx + N_EDGES;

    // ---- workspace carve-out (floats) ----
    float*    ws       = (float*)d_ws;
    float*    Qx       = ws;                                   // N*HD
    float*    Kx       = Qx + (size_t)N_NODES * HDIM;          // N*HD
    float*    Vx       = Kx + (size_t)N_NODES * HDIM;          // N*HD
    float*    scal     = Vx + (size_t)N_NODES * HDIM;          // E*H (scal -> ex)
    unsigned* m_ord    = (unsigned*)(scal + (size_t)N_EDGES * NH);  // N*H
    float*    den      = (float*)(m_ord + (size_t)N_NODES * NH);    // N*H
    float*    node_acc = den + (size_t)N_NODES * NH;           // N*HD
    float*    alpha    = node_acc + (size_t)N_NODES * HDIM;    // N*HD

    float* out_node = (float*)d_out;                           // [N, HD]
    float* edge_rep = out_node + (size_t)N_NODES * HDIM;       // [E, HD]

    // K0: init accumulators
    {
        int total = N_NODES * HDIM;
        grit_init<<<(total + 255) / 256, 256, 0, stream>>>(m_ord, den, node_acc, alpha);
    }
    // K1: node projections (WMMA + TDM-staged weights)
    grit_node_gemm<<<(N_TILES_NODE + 7) / 8, 256, 0, stream>>>(x, WQ, bQ, WK, bK, WV, bV,
                                                               Qx, Kx, Vx);
    // K2: edge projections + elementwise + scores + segment max
    grit_edge_gemm<<<N_EDGES / 128, 256, 0, stream>>>(eattr, WEw, bEw, WEb, bEb,
                                                      srcI, dstI, Qx, Kx, A,
                                                      edge_rep, scal, m_ord);
    // K3: softmax numerator + denominator
    {
        int total = N_EDGES * NH;
        grit_softmax_num<<<(total + 255) / 256, 256, 0, stream>>>(dstI, scal, m_ord, den);
    }
    // K4: scatter-add weighted V and edge_rep
    {
        int total = N_EDGES * HDIM;
        grit_scatter<<<(total + 255) / 256, 256, 0, stream>>>(srcI, dstI, scal, den,
                                                              Vx, edge_rep, node_acc, alpha);
    }
    // K5: final node output with E_v einsum
    {
        int total = N_NODES * HDIM;
        grit_node_out<<<(total + 255) / 256, 256, 0, stream>>>(node_acc, alpha, Ev, out_node);
    }
}